// UncertaintyLoss_60593398612180
// MI455X (gfx1250) — compile-verified
//
#include <hip/hip_runtime.h>
#include <stdint.h>

#define DD 16
#define LFLAT 136                        // 16*17/2
#define GROUP 32                         // items per TDM group (one per lane)
#define NITER 8                          // groups per wave (persistent pipeline)
#define WORDS_PER_GROUP (GROUP * LFLAT)  // 4352 f32 = 17408 B
#define ITEMS_PER_BLOCK (GROUP * NITER)  // 256
#define LOG_2PI 1.8378770664093453f

#if __has_builtin(__builtin_amdgcn_tensor_load_to_lds)
#define USE_TDM 1
#else
#define USE_TDM 0
#endif

typedef unsigned int u32x4 __attribute__((ext_vector_type(4)));
typedef int          i32x8 __attribute__((ext_vector_type(8)));
typedef int          i32x4 __attribute__((ext_vector_type(4)));

// Raw LDS byte offset (address-space(3) view) for a __shared__ pointer.
__device__ __forceinline__ unsigned lds_addr_u32(const void* p) {
  return (unsigned)(unsigned long long)
         (__attribute__((address_space(3))) const void*)p;
}

__device__ __forceinline__ void wait_tensorcnt_le1() {
#if __has_builtin(__builtin_amdgcn_s_wait_tensorcnt)
  __builtin_amdgcn_s_wait_tensorcnt(1);
#else
  asm volatile("s_wait_tensorcnt 0x1" ::: "memory");
#endif
  asm volatile("" ::: "memory");
}

__device__ __forceinline__ void wait_tensorcnt_le0() {
#if __has_builtin(__builtin_amdgcn_s_wait_tensorcnt)
  __builtin_amdgcn_s_wait_tensorcnt(0);
#else
  asm volatile("s_wait_tensorcnt 0x0" ::: "memory");
#endif
  asm volatile("" ::: "memory");
}

#if USE_TDM
// Issue one 1-D TDM copy: GROUP items (17408 B contiguous) -> LDS slice.
__device__ __forceinline__ void issue_tdm(const float* __restrict__ L_flat,
                                          long long item0, int Bn,
                                          unsigned lds0) {
  const unsigned long long gaddr =
      (unsigned long long)(uintptr_t)(L_flat + (size_t)item0 * LFLAT);
  const unsigned long long total_elems = (unsigned long long)Bn * LFLAT;
  const unsigned long long start_elem  = (unsigned long long)item0 * LFLAT;
  const unsigned long long rem =
      (start_elem < total_elems) ? (total_elems - start_elem) : 0ull;
  const unsigned td0 = (rem > 0xffffffffull) ? 0xffffffffu : (unsigned)rem;

  // D# group 0: count=1 | lds_addr | global_addr[56:0] | type=2
  u32x4 g0;
  g0[0] = 1u;
  g0[1] = lds0;
  g0[2] = (unsigned)(gaddr & 0xffffffffu);
  g0[3] = (unsigned)((gaddr >> 32) & 0x01ffffffu) | (2u << 30);

  // D# group 1: data_size=4B; tensor_dim0=td0 (OOB reads return 0);
  // tensor_dim1=1; 1-D tile of 4352 elements; stride0=4352.
  i32x8 g1;
  g1[0] = (int)(2u << 16);                           // data_size = 4 bytes
  g1[1] = (int)((td0 & 0xffffu) << 16);              // tensor_dim0[15:0]
  g1[2] = (int)((td0 >> 16) | (1u << 16));           // tensor_dim0[31:16] | tensor_dim1=1
  g1[3] = (int)(((unsigned)WORDS_PER_GROUP) << 16);  // tile_dim0 = 4352
  g1[4] = 0;                                         // tile_dim1 = tile_dim2 = 0
  g1[5] = (int)WORDS_PER_GROUP;                      // tensor_dim0_stride
  g1[6] = 0;
  g1[7] = 0;
  const i32x4 gz4 = {0, 0, 0, 0};
  const i32x8 gz8 = {0, 0, 0, 0, 0, 0, 0, 0};

  asm volatile("" ::: "memory");
  __builtin_amdgcn_tensor_load_to_lds(g0, g1, gz4, gz4, gz8, 0);
  asm volatile("" ::: "memory");
}
#endif

// Per-item NLL given diff[], mask and the packed lower-triangular rows at Lp.
__device__ __forceinline__ float item_loss(const float* __restrict__ Lp,
                                           const float d[DD], float m,
                                           float invB) {
  float z[DD];
  float logdet = 0.0f, mahal = 0.0f;
  int off = 0;
#pragma unroll
  for (int i = 0; i < DD; ++i) {
    float s = d[i];
#pragma unroll
    for (int j = 0; j < i; ++j) s = fmaf(-Lp[off + j], z[j], s);
    const float dio = Lp[off + i] + 1e-5f;
    const float zi = s / dio;
    z[i] = zi;
    mahal  = fmaf(zi, zi, mahal);
    logdet += __logf(dio);
    off += i + 1;
  }
  const float loss = logdet + 0.5f * (mahal + (float)DD * LOG_2PI);
  return loss * m * invB;
}

__global__ __launch_bounds__(32) void gauss_nll_kernel(
    const float* __restrict__ y_hat,
    const float* __restrict__ L_flat,
    const float* __restrict__ y,
    const float* __restrict__ mask,
    float* __restrict__ block_partials,
    int Bn, float invB)
{
  const int lane = threadIdx.x;  // one wave32 per workgroup
  const long long blockItem0 = (long long)blockIdx.x * ITEMS_PER_BLOCK;

  float partial = 0.0f;

#if USE_TDM
  __shared__ float shL[2 * WORDS_PER_GROUP];  // double buffer, 34816 B

  // Prologue: start DMA for group 0 into buffer 0.
  issue_tdm(L_flat, blockItem0, Bn, lds_addr_u32(&shL[0]));

  for (int g = 0; g < NITER; ++g) {
    const long long item0 = blockItem0 + (long long)g * GROUP;
    const int buf = g & 1;

    // Software pipeline: kick off DMA for the next group into the other
    // buffer before waiting; TENSORcnt completes in order per wave, so
    // s_wait_tensorcnt 1 below guarantees the *current* buffer is ready.
    const bool more = (g + 1) < NITER;
    if (more)
      issue_tdm(L_flat, item0 + GROUP, Bn,
                lds_addr_u32(&shL[(buf ^ 1) * WORDS_PER_GROUP]));

    // Overlap with DMA: pull y / y_hat / mask and form diff.
    const int b = (int)item0 + lane;
    float d[DD];
    float m = 0.0f;
    if (b < Bn) {
      const float4* yv  = reinterpret_cast<const float4*>(y     + (size_t)b * DD);
      const float4* yhv = reinterpret_cast<const float4*>(y_hat + (size_t)b * DD);
#pragma unroll
      for (int k = 0; k < 4; ++k) {
        const float4 a = yv[k], c = yhv[k];
        d[4*k+0] = a.x - c.x; d[4*k+1] = a.y - c.y;
        d[4*k+2] = a.z - c.z; d[4*k+3] = a.w - c.w;
      }
      m = mask[b];
    }

    if (more) wait_tensorcnt_le1();
    else      wait_tensorcnt_le0();

    if (b < Bn) {
      const float* Lp = &shL[buf * WORDS_PER_GROUP + lane * LFLAT];
      partial += item_loss(Lp, d, m, invB);
    }
  }
#else
  // Fallback: compute straight from global memory.
  for (int g = 0; g < NITER; ++g) {
    const int b = (int)(blockItem0 + (long long)g * GROUP) + lane;
    if (b < Bn) {
      float d[DD];
      const float4* yv  = reinterpret_cast<const float4*>(y     + (size_t)b * DD);
      const float4* yhv = reinterpret_cast<const float4*>(y_hat + (size_t)b * DD);
#pragma unroll
      for (int k = 0; k < 4; ++k) {
        const float4 a = yv[k], c = yhv[k];
        d[4*k+0] = a.x - c.x; d[4*k+1] = a.y - c.y;
        d[4*k+2] = a.z - c.z; d[4*k+3] = a.w - c.w;
      }
      partial += item_loss(L_flat + (size_t)b * LFLAT, d, mask[b], invB);
    }
  }
#endif

  // wave32 tree reduction; one partial per workgroup (single wave).
#pragma unroll
  for (int o = 16; o > 0; o >>= 1) partial += __shfl_xor(partial, o, 32);
  if (lane == 0) block_partials[blockIdx.x] = partial;
}

__global__ __launch_bounds__(256) void reduce_partials_kernel(
    const float* __restrict__ partials, int n, float* __restrict__ out)
{
  __shared__ float sred[8];
  float v = 0.0f;
  for (int i = threadIdx.x; i < n; i += 256) v += partials[i];
#pragma unroll
  for (int o = 16; o > 0; o >>= 1) v += __shfl_xor(v, o, 32);
  const int lane = threadIdx.x & 31, wave = threadIdx.x >> 5;
  if (lane == 0) sred[wave] = v;
  __syncthreads();
  if (threadIdx.x == 0) {
    float s = 0.0f;
#pragma unroll
    for (int w = 0; w < 8; ++w) s += sred[w];
    out[0] = s;
  }
}

extern "C" void kernel_launch(void* const* d_in, const int* in_sizes, int n_in,
                              void* d_out, int out_size, void* d_ws, size_t ws_size,
                              hipStream_t stream) {
  const float* y_hat  = (const float*)d_in[0];
  const float* L_flat = (const float*)d_in[1];
  const float* y      = (const float*)d_in[2];
  const float* mask   = (const float*)d_in[3];
  const int Bn = in_sizes[0] / DD;
  const int nblocks = (Bn + ITEMS_PER_BLOCK - 1) / ITEMS_PER_BLOCK;
  const float invB = 1.0f / (float)Bn;

  gauss_nll_kernel<<<nblocks, 32, 0, stream>>>(
      y_hat, L_flat, y, mask, (float*)d_ws, Bn, invB);
  reduce_partials_kernel<<<1, 256, 0, stream>>>(
      (const float*)d_ws, nblocks, (float*)d_out);
}